// WARP_63599875719298
// MI455X (gfx1250) — compile-verified
//
#include <hip/hip_runtime.h>
#include <hip/hip_bf16.h>

// Problem constants (match reference)
#define BATCH   32
#define LSEQ    32
#define XDIM    9
#define IN_DIM  10
#define DTH     4609           // D_THETA
#define KFULL   4608           // DTH rounded down to multiple of 4 (WMMA K chunks)
#define NTILES  289            // ceil(4609/16)
#define KSPLIT  8              // waves per block, each owns KFULL/KSPLIT = 576 of K
#define KCHUNK  (KFULL / KSPLIT)
#define WAVES_PER_BLOCK 8

typedef __attribute__((ext_vector_type(2))) float v2f;
typedef __attribute__((ext_vector_type(8))) float v8f;

// ---------------------------------------------------------------------------
// Kernel 1: build u = concat(ts, xs) and du = diff(u, prepend=u[:, :1])
// ---------------------------------------------------------------------------
__global__ void WARP_prep_kernel(const float* __restrict__ xs,
                                 const float* __restrict__ ts,
                                 float* __restrict__ u,
                                 float* __restrict__ du) {
    int idx = blockIdx.x * blockDim.x + threadIdx.x;
    if (idx >= BATCH * LSEQ * IN_DIM) return;
    int i = idx % IN_DIM;          // feature
    int t = idx / IN_DIM;          // flat (b*L + l)
    int l = t % LSEQ;
    float cur = (i == 0) ? ts[t] : xs[t * XDIM + (i - 1)];
    float prev = cur;              // l==0 -> du = 0
    if (l > 0) {
        int tp = t - 1;
        prev = (i == 0) ? ts[tp] : xs[tp * XDIM + (i - 1)];
    }
    u[idx]  = cur;
    du[idx] = cur - prev;
}

// ---------------------------------------------------------------------------
// Kernel 2 (launched LSEQ times): one scan step.
//   theta_out[m, n] = sum_k theta_prev[m,k]*A[n,k] + sum_i du[m,i]*Bw[n,i]
// M=32 (batch), N=K=4609.
// One BLOCK per 32x16 output tile; 8 waves split K (576 each) with f32 WMMA
// (16x16, K=4), partials combined via LDS in fixed order (deterministic).
// prev_stride==0 broadcasts theta_0 across the batch for step 0.
// ---------------------------------------------------------------------------
__global__ __launch_bounds__(256)
void WARP_step_kernel(const float* __restrict__ Amat,      // (DTH, DTH) row-major
                      const float* __restrict__ Bw,        // (DTH, IN_DIM)
                      const float* __restrict__ theta_prev,// row m at m*prev_stride
                      int prev_stride,
                      const float* __restrict__ du_l,      // row m at m*(L*IN_DIM)
                      float* __restrict__ theta_out) {     // row m at m*(L*DTH)
    __shared__ float red[KSPLIT][16][32];                  // 16 KB

    const int wave = threadIdx.x >> 5;
    const int lane = threadIdx.x & 31;
    const int tile = blockIdx.x;                           // < NTILES always

    const int nlo   = lane & 15;
    const int khalf = lane >> 4;           // 0: k0/k0+1, 1: k0+2/k0+3
    const int koff  = khalf * 2;
    const int n     = tile * 16 + nlo;
    const int nc    = (n < DTH) ? n : (DTH - 1);   // clamp loads, guard stores

    // B-operand: column n of A^T == memory row n of A (contiguous in k).
    const float* __restrict__ Brow = Amat + (size_t)nc * DTH;
    // A-operand rows (theta) for the two 16-row M tiles (ISA 7.12.2 layout).
    const float* __restrict__ th0 = theta_prev + (size_t)nlo        * prev_stride;
    const float* __restrict__ th1 = theta_prev + (size_t)(nlo + 16) * prev_stride;

    v8f acc0 = {};
    v8f acc1 = {};

    const int kbeg = wave * KCHUNK;
    const int kend = kbeg + KCHUNK;        // 144 chunks of K=4 per wave
#pragma unroll 4
    for (int k0 = kbeg; k0 < kend; k0 += 4) {
        v2f b  = *(const v2f*)(Brow + k0 + koff);
        v2f a0 = *(const v2f*)(th0  + k0 + koff);
        v2f a1 = *(const v2f*)(th1  + k0 + koff);
        acc0 = __builtin_amdgcn_wmma_f32_16x16x4_f32(false, a0, false, b,
                                                     (short)0, acc0, false, false);
        acc1 = __builtin_amdgcn_wmma_f32_16x16x4_f32(false, a1, false, b,
                                                     (short)0, acc1, false, false);
    }

    // Stash per-wave partials; lanes write consecutive addresses (no conflicts).
#pragma unroll
    for (int v = 0; v < 8; ++v) {
        red[wave][v][lane]     = acc0[v];
        red[wave][8 + v][lane] = acc1[v];
    }
    __syncthreads();

    // Reduction + epilogue. Thread (wave==v, lane) owns both M-halves of its
    // (v, lane) slot: m0 = v + 8*khalf (tile M 0..15), m1 = m0 + 16.
    const int v = wave;
    float s0 = 0.f, s1 = 0.f;
#pragma unroll
    for (int w = 0; w < KSPLIT; ++w) {     // fixed order -> deterministic
        s0 += red[w][v][lane];
        s1 += red[w][8 + v][lane];
    }

    // K remainder (k = 4608) + du @ Bw^T bias, shared per-n loads.
    const float a_rem = Brow[KFULL];       // A[n, 4608]
    float bw[IN_DIM];
#pragma unroll
    for (int i = 0; i < IN_DIM; ++i) bw[i] = Bw[(size_t)nc * IN_DIM + i];

    const int m0 = v + 8 * khalf;
    const int m1 = m0 + 16;
    float val0 = s0 + theta_prev[(size_t)m0 * prev_stride + KFULL] * a_rem;
    float val1 = s1 + theta_prev[(size_t)m1 * prev_stride + KFULL] * a_rem;

    const float* du0 = du_l + (size_t)m0 * (LSEQ * IN_DIM);
    const float* du1 = du_l + (size_t)m1 * (LSEQ * IN_DIM);
    float d0 = 0.f, d1 = 0.f;
#pragma unroll
    for (int i = 0; i < IN_DIM; ++i) {
        d0 += du0[i] * bw[i];
        d1 += du1[i] * bw[i];
    }
    val0 += d0;
    val1 += d1;

    if (n < DTH) {                          // only tile 288 / n==4608 edge
        theta_out[(size_t)m0 * (LSEQ * DTH) + n] = val0;
        theta_out[(size_t)m1 * (LSEQ * DTH) + n] = val1;
    }
}

// ---------------------------------------------------------------------------
// Kernel 3: per-token hypernetwork MLP. One wave per token; lane o owns output
// neuron o (hidden width 32 == wave32). Activations broadcast via __shfl.
// theta layout: W0[320] b0[32] | 4x (W[1024] b[32]) | W5[32] b5[1]
// ---------------------------------------------------------------------------
__global__ __launch_bounds__(256)
void WARP_mlp_kernel(const float* __restrict__ ws_theta,
                     const float* __restrict__ ws_u,
                     float* __restrict__ out) {
    const int wave = threadIdx.x >> 5;
    const int lane = threadIdx.x & 31;
    const int t = blockIdx.x * WAVES_PER_BLOCK + wave;   // token (b*L + l), < 1024
    const float* __restrict__ th = ws_theta + (size_t)t * DTH;
    const float* __restrict__ u  = ws_u + (size_t)t * IN_DIM;

    float x[IN_DIM];
#pragma unroll
    for (int i = 0; i < IN_DIM; ++i) x[i] = u[i];

    // Layer 0: (32, 10)
    float acc = th[320 + lane];
#pragma unroll
    for (int i = 0; i < IN_DIM; ++i) acc += th[lane * IN_DIM + i] * x[i];
    float h = fmaxf(acc, 0.f);

    // Hidden layers 1..4: (32, 32)
    int off = 352;
    for (int layer = 0; layer < 4; ++layer) {
        const float* __restrict__ W = th + off;
        float a2 = th[off + 1024 + lane];
#pragma unroll
        for (int i = 0; i < 32; ++i)
            a2 += W[lane * 32 + i] * __shfl(h, i, 32);
        h = fmaxf(a2, 0.f);
        off += 1056;
    }

    // Output layer: (1, 32) -> wave reduction
    float p = th[4576 + lane] * h;
#pragma unroll
    for (int s = 16; s > 0; s >>= 1) p += __shfl_xor(p, s, 32);
    if (lane == 0) out[t] = p + th[4608];
}

// ---------------------------------------------------------------------------
// Launch: prep, 32 sequential scan steps (linear recurrence), MLP readout.
// Workspace: u (40KB) | du (40KB) | theta_outs (B*L*DTH f32 ~= 18.9MB)
// ---------------------------------------------------------------------------
extern "C" void kernel_launch(void* const* d_in, const int* in_sizes, int n_in,
                              void* d_out, int out_size, void* d_ws, size_t ws_size,
                              hipStream_t stream) {
    const float* xs  = (const float*)d_in[0];   // (B, L, 9)
    const float* ts  = (const float*)d_in[1];   // (B, L, 1)
    const float* A   = (const float*)d_in[2];   // (DTH, DTH)
    const float* Bw  = (const float*)d_in[3];   // (DTH, IN_DIM)
    const float* th0 = (const float*)d_in[4];   // (DTH,)
    float* out = (float*)d_out;                 // (B, L, 1) -> 1024 floats

    float* ws_u     = (float*)d_ws;
    float* ws_du    = ws_u  + (size_t)BATCH * LSEQ * IN_DIM;
    float* ws_theta = ws_du + (size_t)BATCH * LSEQ * IN_DIM;  // (B, L, DTH)

    const int nprep = BATCH * LSEQ * IN_DIM;
    WARP_prep_kernel<<<(nprep + 255) / 256, 256, 0, stream>>>(xs, ts, ws_u, ws_du);

    for (int l = 0; l < LSEQ; ++l) {
        const float* prev = (l == 0) ? th0 : (ws_theta + (size_t)(l - 1) * DTH);
        const int pstride = (l == 0) ? 0 : (LSEQ * DTH);  // 0 broadcasts theta_0
        WARP_step_kernel<<<NTILES, 256, 0, stream>>>(
            A, Bw, prev, pstride,
            ws_du + (size_t)l * IN_DIM,
            ws_theta + (size_t)l * DTH);
    }

    WARP_mlp_kernel<<<(BATCH * LSEQ) / WAVES_PER_BLOCK, 256, 0, stream>>>(
        ws_theta, ws_u, out);
}